// BSplineBasis_90829968376333
// MI455X (gfx1250) — compile-verified
//
#include <hip/hip_runtime.h>
#include <hip/hip_bf16.h>
#include <cstdint>

// B-spline basis (degree 3, 10 knots -> 6 bases) over 8M points.
// Memory-bound: ~256MB traffic -> ~11us roofline on MI455X (23.3 TB/s).
// CDNA5 path: b128 vector loads, LDS staging, async LDS->global b128 stores
// (ASYNCcnt, non-temporal) so each wave store instruction writes a contiguous
// 512B segment and the 192MB output stream does not thrash the 192MB L2.

#define NKNOTS 10
#define DEG    3
#define NBASES (NKNOTS - DEG - 1)   // 6
#define BLK    256
#define PPT    4                    // points per thread
#define PPB    (BLK * PPT)          // 1024 points per block
#define STAGE_WORDS (PPB * NBASES)  // 6144 floats = 24KB LDS

typedef float f32x4 __attribute__((ext_vector_type(4)));  // clang vector: OK for NT builtins

// ---- order-preserving float<->uint encoding for integer atomics ----
__device__ __forceinline__ unsigned fenc(float f) {
    unsigned u = __float_as_uint(f);
    return (u & 0x80000000u) ? ~u : (u | 0x80000000u);
}
__device__ __forceinline__ float fdec(unsigned e) {
    unsigned u = (e & 0x80000000u) ? (e & 0x7FFFFFFFu) : ~e;
    return __uint_as_float(u);
}

__global__ void bspl_init_mm(unsigned* mm) {
    mm[0] = 0xFFFFFFFFu;  // running min (encoded)
    mm[1] = 0x00000000u;  // running max (encoded)
}

__global__ __launch_bounds__(BLK) void bspl_minmax(const float* __restrict__ x,
                                                   int n, unsigned* __restrict__ mm) {
    __shared__ float slo[BLK];
    __shared__ float shi[BLK];
    const int tid   = threadIdx.x;
    const int gtid  = blockIdx.x * BLK + tid;
    const int total = gridDim.x * BLK;

    float lo =  __builtin_inff();
    float hi = -__builtin_inff();

    const int n4 = n >> 2;
    const f32x4* x4 = reinterpret_cast<const f32x4*>(x);
    // Default (RT) loads on purpose: this warms L2 with x for the eval pass.
    for (int i = gtid; i < n4; i += total) {
        f32x4 v = x4[i];
        lo = fminf(lo, fminf(fminf(v.x, v.y), fminf(v.z, v.w)));
        hi = fmaxf(hi, fmaxf(fmaxf(v.x, v.y), fmaxf(v.z, v.w)));
    }
    for (int i = (n4 << 2) + gtid; i < n; i += total) {
        float v = x[i];
        lo = fminf(lo, v);
        hi = fmaxf(hi, v);
    }

    slo[tid] = lo; shi[tid] = hi;
    __syncthreads();
    for (int s = BLK >> 1; s > 0; s >>= 1) {
        if (tid < s) {
            slo[tid] = fminf(slo[tid], slo[tid + s]);
            shi[tid] = fmaxf(shi[tid], shi[tid + s]);
        }
        __syncthreads();
    }
    if (tid == 0) {
        atomicMin(&mm[0], fenc(slo[0]));
        atomicMax(&mm[1], fenc(shi[0]));
    }
}

__global__ __launch_bounds__(BLK) void bspl_eval(const float* __restrict__ x,
                                                 const float* __restrict__ knots,
                                                 const unsigned* __restrict__ mm,
                                                 float* __restrict__ out, int n) {
    __shared__ float stage[STAGE_WORDS];  // mirrors global output layout of this block

    const int tid = threadIdx.x;
    const long long blockStart = (long long)blockIdx.x * PPB;
    const bool full = (blockStart + PPB) <= (long long)n;

    // normalization constants (uniform)
    const float tmin = fdec(mm[0]);
    const float tmax = fdec(mm[1]);
    const float inv_range = 1.0f / (tmax - tmin + 1e-8f);

    // knots into registers (uniform loads, cached)
    float t[NKNOTS];
#pragma unroll
    for (int i = 0; i < NKNOTS; ++i) t[i] = knots[i];

    // Precompute uniform reciprocal denominators for all 21 weight pairs.
    // inv == 0 exactly reproduces the reference's where(den>0, ., 0).
    float i1[21], i2[21];
    {
        int k = 0;
#pragma unroll
        for (int d = 1; d <= DEG; ++d) {
#pragma unroll
            for (int i = 0; i < NKNOTS - 1 - d; ++i) {
                float den1 = t[i + d] - t[i];
                float den2 = t[i + d + 1] - t[i + 1];
                i1[k] = (den1 > 0.0f) ? (1.0f / den1) : 0.0f;
                i2[k] = (den2 > 0.0f) ? (1.0f / den2) : 0.0f;
                ++k;
            }
        }
    }

    // ---- load 4 points (last use of x -> non-temporal) ----
    const int base = tid * PPT;                 // local point index base
    const long long gp = blockStart + base;     // global point index base
    float xs[PPT];
    if (full) {
        const f32x4* xv4 = reinterpret_cast<const f32x4*>(x) + (blockStart >> 2) + tid;
        f32x4 v = __builtin_nontemporal_load(xv4);
        xs[0] = v.x; xs[1] = v.y; xs[2] = v.z; xs[3] = v.w;
    } else {
#pragma unroll
        for (int p = 0; p < PPT; ++p)
            xs[p] = (gp + p < n) ? x[gp + p] : 0.0f;
    }

    // ---- Cox-de Boor per point, fully unrolled ----
#pragma unroll
    for (int p = 0; p < PPT; ++p) {
        float xn = (xs[p] - tmin) * inv_range;

        float B[NKNOTS - 1];
#pragma unroll
        for (int i = 0; i < NKNOTS - 1; ++i) {
            float b = (xn >= t[i] && xn < t[i + 1]) ? 1.0f : 0.0f;
            // clamped right-endpoint rule
            if (xn == t[NKNOTS - 1] && t[i + 1] == t[NKNOTS - 1] && t[i] < t[i + 1])
                b = 1.0f;
            B[i] = b;
        }
        int k = 0;
#pragma unroll
        for (int d = 1; d <= DEG; ++d) {
#pragma unroll
            for (int i = 0; i < NKNOTS - 1 - d; ++i) {
                float w1 = (xn - t[i]) * i1[k];
                float w2 = (t[i + d + 1] - xn) * i2[k];
                B[i] = w1 * B[i] + w2 * B[i + 1];
                ++k;
            }
        }
#pragma unroll
        for (int j = 0; j < NBASES; ++j)
            stage[(base + p) * NBASES + j] = B[j];
    }

    __syncthreads();  // all LDS staging visible (waits DScnt + wg barrier)

    if (full) {
        // Async LDS -> global streaming stores: 24KB per block, 6 chunks of
        // 16B per thread; each wave instruction covers a contiguous 512B.
        // Non-temporal: 192MB output stream must not evict L2-resident x.
        const unsigned ldsBase = (unsigned)(uintptr_t)(&stage[0]);
        const unsigned long long gbase =
            (unsigned long long)(uintptr_t)(out + blockStart * NBASES);
        const int CHUNKS = (STAGE_WORDS * 4) / 16 / BLK;  // 6
#pragma unroll
        for (int c = 0; c < CHUNKS; ++c) {
            unsigned off = (unsigned)(tid + c * BLK) * 16u;
            unsigned lds_addr = ldsBase + off;
            unsigned long long gaddr = gbase + off;
            asm volatile("global_store_async_from_lds_b128 %0, %1, off th:TH_STORE_NT"
                         :: "v"(gaddr), "v"(lds_addr)
                         : "memory");
        }
        asm volatile("s_wait_asynccnt 0" ::: "memory");
    } else {
        // tail block: guarded direct stores
#pragma unroll
        for (int p = 0; p < PPT; ++p) {
            long long gi = gp + p;
            if (gi < n) {
#pragma unroll
                for (int j = 0; j < NBASES; ++j)
                    out[gi * NBASES + j] = stage[(base + p) * NBASES + j];
            }
        }
    }
}

extern "C" void kernel_launch(void* const* d_in, const int* in_sizes, int n_in,
                              void* d_out, int out_size, void* d_ws, size_t ws_size,
                              hipStream_t stream) {
    const float* x     = (const float*)d_in[0];
    const float* knots = (const float*)d_in[1];
    const int n = in_sizes[0];
    float* out = (float*)d_out;
    unsigned* mm = (unsigned*)d_ws;

    bspl_init_mm<<<1, 1, 0, stream>>>(mm);

    int rblocks = 2048;
    bspl_minmax<<<rblocks, BLK, 0, stream>>>(x, n, mm);

    int eblocks = (int)((n + PPB - 1) / PPB);
    bspl_eval<<<eblocks, BLK, 0, stream>>>(x, knots, mm, out, n);
}